// LinearCRF_29557964931770
// MI455X (gfx1250) — compile-verified
//
#include <hip/hip_runtime.h>
#include <stdint.h>

// Problem constants (from reference setup_inputs: B=256, T=1024, L=128).
static constexpr int cB = 256;
static constexpr int cT = 1024;
static constexpr int cL = 128;
static constexpr int TRS = 132;  // padded LDS row stride (floats); 16B-aligned rows

// ---------------------------------------------------------------------------
// CDNA5 async global->LDS copy (gfx1250). Tracked with ASYNCcnt.
// GV mode: vdst = LDS byte address VGPR, vaddr = 64-bit global address pair.
// ---------------------------------------------------------------------------
__device__ __forceinline__ void async_ld_b32(unsigned lds_addr, const void* gaddr) {
  asm volatile("global_load_async_to_lds_b32 %0, %1, off"
               :: "v"(lds_addr), "v"(gaddr)
               : "memory");
}
__device__ __forceinline__ void wait_async_le1(void) {
  asm volatile("s_wait_asynccnt 1" ::: "memory");
}
__device__ __forceinline__ void wait_async_0(void) {
  asm volatile("s_wait_asynccnt 0" ::: "memory");
}
__device__ __forceinline__ unsigned lds_off32(const void* p) {
  // Generic pointer to LDS: low 32 bits are the LDS byte offset (ISA 10.2).
  return (unsigned)(size_t)p;
}

// ---------------------------------------------------------------------------
// One block per batch element; thread j owns tag j (128 threads = 4 wave32s).
// Forward (max-plus, WMMA inapplicable): transition row lives in 128 VGPRs
// per lane; shared score vector is broadcast from LDS (uniform-address
// ds_load_b128). All forward scores stream to d_ws (128 MB, fits the 192 MB
// L2); backtrace recomputes one argmax per step from L2-resident scores,
// with the tag-independent score load software-pipelined one step ahead.
// ---------------------------------------------------------------------------
__global__ __launch_bounds__(cL)
void viterbi_crf_kernel(const float* __restrict__ x,
                        const unsigned char* __restrict__ mask,  // JAX bool = 1 byte
                        const float* __restrict__ tr,
                        float* __restrict__ out,   // [B*T path as float][B score]
                        float* __restrict__ S)     // d_ws: B*T*L forward scores
{
  __shared__ float trT[cL * TRS];   // trT[j][i] = tr[i][j] (kept for backtrace)
  __shared__ float sbuf[2 * cL];    // ping-pong scores
  __shared__ float emis[2 * cL];    // async-filled emission double buffer
  __shared__ float red[2];          // final (score, argmax-as-bits)

  const int j = threadIdx.x;
  const int b = blockIdx.x;
  const float* xb = x + (size_t)b * cT * cL;
  float* Sb = S + (size_t)b * cT * cL;
  const unsigned char* mb = mask + (size_t)b * cT;

  // Transpose transition into LDS (coalesced across threads per i).
  for (int i = 0; i < cL; ++i)
    trT[j * TRS + i] = tr[i * cL + j];

  // t = 0: init scores = x[:,0]
  float myscore = xb[j];
  sbuf[j] = myscore;
  Sb[j] = myscore;

  // Prefetch emission for t=1 via async global->LDS (each lane: own element).
  async_ld_b32(lds_off32(&emis[cL + j]), &xb[(size_t)cL + j]);
  __syncthreads();

  // Hoist this lane's transition row into registers (loop-invariant over t).
  float treg[cL];
#pragma unroll
  for (int i = 0; i < cL; i += 4) {
    float4 tv = *(const float4*)&trT[j * TRS + i];
    treg[i]     = tv.x;
    treg[i + 1] = tv.y;
    treg[i + 2] = tv.z;
    treg[i + 3] = tv.w;
  }

  int cur = 0;
  const float NEG = -3.402823466e38f;

  for (int t = 1; t < cT; ++t) {
    // Pipeline: kick off next step's emission while we crunch this one.
    if (t + 1 < cT)
      async_ld_b32(lds_off32(&emis[((t + 1) & 1) * cL + j]),
                   &xb[(size_t)(t + 1) * cL + j]);

    const float* sc = &sbuf[cur * cL];
    float b0 = NEG, b1 = NEG, b2 = NEG, b3 = NEG;
#pragma unroll
    for (int i = 0; i < cL; i += 4) {
      float4 sv = *(const float4*)&sc[i];   // uniform-address LDS broadcast
      b0 = fmaxf(b0, sv.x + treg[i]);
      b1 = fmaxf(b1, sv.y + treg[i + 1]);
      b2 = fmaxf(b2, sv.z + treg[i + 2]);
      b3 = fmaxf(b3, sv.w + treg[i + 3]);
    }
    float best = fmaxf(fmaxf(b0, b1), fmaxf(b2, b3));

    // Make this step's async emission visible (in-order completion; only the
    // t+1 prefetch may still be outstanding).
    if (t + 1 < cT) wait_async_le1(); else wait_async_0();
    float e = emis[(t & 1) * cL + j];

    float nv = mb[t] ? (best + e) : myscore;   // reference mask semantics
    myscore = nv;
    int nxt = cur ^ 1;
    sbuf[nxt * cL + j] = nv;
    Sb[(size_t)t * cL + j] = nv;               // stream to L2-resident d_ws
    __syncthreads();
    cur = nxt;
  }

  // Final max/argmax over 128 tags (first-max index, matching jnp.argmax).
  if (j == 0) {
    const float* fs = &sbuf[cur * cL];
    float bestv = fs[0]; int bi = 0;
    for (int i = 1; i < cL; ++i) { float v = fs[i]; if (v > bestv) { bestv = v; bi = i; } }
    red[0] = bestv;
    red[1] = __int_as_float(bi);
    out[(size_t)cB * cT + b] = bestv;                // score[b]
    out[(size_t)b * cT + (cT - 1)] = (float)bi;      // path[b][T-1]
  }
  __threadfence();   // make Sb stores visible to our own backtrace loads
  __syncthreads();

  // Backtrace on wave 0: recompute bp_t[tag] = argmax_i(S[t-1][i]+tr[i][tag]).
  // The S-row load address depends only on t (not the loop-carried tag), so
  // prefetch it one iteration ahead to hide L2 latency behind the reduction.
  if (j < 32) {
    int tag = __float_as_int(red[1]);
    float4 sv = *(const float4*)&Sb[(size_t)(cT - 2) * cL + j * 4];
    for (int t = cT - 1; t >= 1; --t) {
      float4 svn;
      if (t >= 2)   // issue next row's load before the dependent argmax chain
        svn = *(const float4*)&Sb[(size_t)(t - 2) * cL + j * 4];
      int prev;
      if (mb[t]) {
        float4 tv = *(const float4*)&trT[tag * TRS + j * 4];
        float v0 = sv.x + tv.x, v1 = sv.y + tv.y;
        float v2 = sv.z + tv.z, v3 = sv.w + tv.w;
        float bv = v0; int bi = j * 4;
        if (v1 > bv) { bv = v1; bi = j * 4 + 1; }
        if (v2 > bv) { bv = v2; bi = j * 4 + 2; }
        if (v3 > bv) { bv = v3; bi = j * 4 + 3; }
        // wave32 shuffle reduce; equal values resolve to the smaller index,
        // which preserves first-max tie-breaking.
        for (int off = 16; off > 0; off >>= 1) {
          float ov = __shfl_down(bv, off);
          int   oi = __shfl_down(bi, off);
          if (ov > bv || (ov == bv && oi < bi)) { bv = ov; bi = oi; }
        }
        prev = __shfl(bi, 0);
      } else {
        prev = tag;   // masked step: identity backpointer
      }
      if (j == 0) out[(size_t)b * cT + (t - 1)] = (float)prev;
      tag = prev;
      sv = svn;
    }
  }
}

extern "C" void kernel_launch(void* const* d_in, const int* in_sizes, int n_in,
                              void* d_out, int out_size, void* d_ws, size_t ws_size,
                              hipStream_t stream) {
  const float* x = (const float*)d_in[0];                    // (B,T,L) f32
  const unsigned char* mask = (const unsigned char*)d_in[1]; // (B,T) bool
  const float* tr = (const float*)d_in[2];                   // (L,L) f32
  float* out = (float*)d_out;                                // path(B*T) ++ score(B)
  float* S = (float*)d_ws;                                   // B*T*L f32 = 128 MB

  viterbi_crf_kernel<<<dim3(cB), dim3(cL), 0, stream>>>(x, mask, tr, out, S);
}